// GNNGI_52295521796228
// MI455X (gfx1250) — compile-verified
//
#include <hip/hip_runtime.h>
#include <hip/hip_bf16.h>
#include <math.h>

#define N_NODES 50000
#define E_EDGES 400000
#define NG_     50
#define NPG     1000      // nodes per graph (contiguous blocks)
#define H_      128
#define L_      3
#define PSI_IN  193
#define PHI_IN  256
#define EPS_    1e-5f
#define NTILES  3125      // 50000 / 16
#define NSCAN   196       // ceil(50000/256)

typedef float v2f __attribute__((ext_vector_type(2)));
typedef float v8f __attribute__((ext_vector_type(8)));

// ---------------------------------------------------------------------------
// f32 WMMA 16x16x4:  D = A(16x4) * B(4x16) + C   (full f32 precision)
// ---------------------------------------------------------------------------
__device__ __forceinline__ v8f wmma4(v2f a, v2f b, v8f c) {
  return __builtin_amdgcn_wmma_f32_16x16x4_f32(
      /*neg_a=*/false, a, /*neg_b=*/false, b,
      /*c_mod=*/(short)0, c, /*reuse_a=*/false, /*reuse_b=*/false);
}

// One wave accumulates a 16x64 strip of C = A@B.  (layouts per ISA 7.12.2)
__device__ __forceinline__ void gemm_strip64(const float* __restrict__ arow,
                                             const float* __restrict__ B, int ldb,
                                             int K, int koff, int n,
                                             v8f& c0, v8f& c1, v8f& c2, v8f& c3) {
  for (int k = 0; k < K; k += 4) {
    int kk = k + koff;
    v2f a; a.x = arow[kk]; a.y = arow[kk + 1];
    const float* bp = B + (size_t)kk * ldb + n;
    v2f b0, b1, b2, b3;
    b0.x = bp[0];   b0.y = bp[ldb];
    b1.x = bp[16];  b1.y = bp[ldb + 16];
    b2.x = bp[32];  b2.y = bp[ldb + 32];
    b3.x = bp[48];  b3.y = bp[ldb + 48];
    c0 = wmma4(a, b0, c0);
    c1 = wmma4(a, b1, c1);
    c2 = wmma4(a, b2, c2);
    c3 = wmma4(a, b3, c3);
  }
}

// ---------------------------------------------------------------------------
// edge_attr mean/var (two-pass deterministic reduction)
// ---------------------------------------------------------------------------
__global__ void k_stat1(const float* __restrict__ ea, float* __restrict__ part) {
  __shared__ float ss[256], sq[256];
  float s = 0.f, q = 0.f;
  for (int i = blockIdx.x * blockDim.x + threadIdx.x; i < E_EDGES;
       i += gridDim.x * blockDim.x) {
    float v = ea[i];
    s += v; q += v * v;
  }
  ss[threadIdx.x] = s; sq[threadIdx.x] = q;
  __syncthreads();
  for (int o = 128; o > 0; o >>= 1) {
    if (threadIdx.x < o) {
      ss[threadIdx.x] += ss[threadIdx.x + o];
      sq[threadIdx.x] += sq[threadIdx.x + o];
    }
    __syncthreads();
  }
  if (threadIdx.x == 0) {
    part[2 * blockIdx.x]     = ss[0];
    part[2 * blockIdx.x + 1] = sq[0];
  }
}

__global__ void k_stat2(const float* __restrict__ part, float* __restrict__ stats) {
  __shared__ float ss[256], sq[256];
  ss[threadIdx.x] = part[2 * threadIdx.x];
  sq[threadIdx.x] = part[2 * threadIdx.x + 1];
  __syncthreads();
  for (int o = 128; o > 0; o >>= 1) {
    if (threadIdx.x < o) {
      ss[threadIdx.x] += ss[threadIdx.x + o];
      sq[threadIdx.x] += sq[threadIdx.x + o];
    }
    __syncthreads();
  }
  if (threadIdx.x == 0) {
    float mu  = ss[0] / (float)E_EDGES;
    float var = sq[0] / (float)E_EDGES - mu * mu;
    stats[0] = mu;
    stats[1] = rsqrtf(var + EPS_);
  }
}

// ---------------------------------------------------------------------------
// CSR build (once): count -> scan(3 kernels) -> fill buckets
// ---------------------------------------------------------------------------
__global__ void k_count(const int* __restrict__ dst, int* __restrict__ deg_i) {
  int e = blockIdx.x * blockDim.x + threadIdx.x;
  if (e >= E_EDGES) return;
  atomicAdd(&deg_i[dst[e]], 1);
}

__global__ void k_scan1(const int* __restrict__ deg_i, int* __restrict__ off,
                        int* __restrict__ bsum) {
  __shared__ int sm[256];
  int t = threadIdx.x;
  int n = blockIdx.x * 256 + t;
  int v = (n < N_NODES) ? deg_i[n] : 0;
  sm[t] = v;
  __syncthreads();
  for (int o = 1; o < 256; o <<= 1) {
    int add = (t >= o) ? sm[t - o] : 0;
    __syncthreads();
    sm[t] += add;
    __syncthreads();
  }
  if (n < N_NODES) off[n] = sm[t] - v;          // exclusive within block
  if (t == 255) bsum[blockIdx.x] = sm[255];     // block total
}

__global__ void k_scan2(int* __restrict__ bsum) {   // single block, in-place excl scan
  __shared__ int sm[256];
  int t = threadIdx.x;
  int v = (t < NSCAN) ? bsum[t] : 0;
  sm[t] = v;
  __syncthreads();
  for (int o = 1; o < 256; o <<= 1) {
    int add = (t >= o) ? sm[t - o] : 0;
    __syncthreads();
    sm[t] += add;
    __syncthreads();
  }
  if (t < NSCAN) bsum[t] = sm[t] - v;
}

__global__ void k_scan3(int* __restrict__ off, const int* __restrict__ bsum) {
  int n = blockIdx.x * blockDim.x + threadIdx.x;
  if (n < N_NODES) off[n] += bsum[n >> 8];
}

// bucket fill: per edge store src node and normalized edge attr
__global__ void k_fill(const int* __restrict__ src, const int* __restrict__ dst,
                       const float* __restrict__ ea, const float* __restrict__ stats,
                       const float* __restrict__ gamma, const float* __restrict__ beta,
                       const int* __restrict__ off, int* __restrict__ cursor,
                       int* __restrict__ elist, float* __restrict__ eaval) {
  int e = blockIdx.x * blockDim.x + threadIdx.x;
  if (e >= E_EDGES) return;
  int d = dst[e];
  int slot = off[d] + atomicAdd(&cursor[d], 1);
  elist[slot] = src[e];
  eaval[slot] = (ea[e] - stats[0]) * stats[1] * gamma[0] + beta[0];
}

// ---------------------------------------------------------------------------
// x = concat(emb_table[station_ids], x_feats)  -> N x 64
// ---------------------------------------------------------------------------
__global__ void k_buildx(const float* __restrict__ xf, const int* __restrict__ st,
                         const float* __restrict__ emb, float* __restrict__ x) {
  int t = blockIdx.x * blockDim.x + threadIdx.x;
  if (t >= N_NODES * 64) return;
  int n = t >> 6, f = t & 63;
  x[t] = (f < 16) ? emb[st[n] * 16 + f] : xf[n * 48 + (f - 16)];
}

// gf0[g] = mean over 1000 contiguous nodes
__global__ void k_gf0(const float* __restrict__ x, float* __restrict__ gf0) {
  __shared__ float sm[256];
  int g = blockIdx.x;
  int f = threadIdx.x & 63;
  int part = threadIdx.x >> 6;
  float s = 0.f;
  for (int i = part; i < NPG; i += 4)
    s += x[(size_t)(g * NPG + i) * 64 + f];
  sm[threadIdx.x] = s;
  __syncthreads();
  if (part == 0)
    gf0[g * 64 + f] = (sm[f] + sm[64 + f] + sm[128 + f] + sm[192 + f]) * (1.0f / NPG);
}

// P[v, 0:128]  = x[v] @ W_dst  (psi rows 0..63)
// P[v,128:256] = x[v] @ W_src  (psi rows 64..127)        [WMMA]
__global__ void k_p12(const float* __restrict__ x, const float* __restrict__ psiWl,
                      float* __restrict__ P) {
  int wid  = (blockIdx.x * blockDim.x + threadIdx.x) >> 5;
  int lane = threadIdx.x & 31;
  if (wid >= NTILES * 4) return;
  int rt = wid >> 2;
  int g  = wid & 3;           // 0,1 -> W_dst cols ; 2,3 -> W_src cols
  int row0 = rt * 16;
  const float* B = psiWl + (g >= 2 ? (size_t)64 * H_ : 0) + (size_t)(g & 1) * 64;
  int n    = lane & 15;
  int koff = (lane >> 4) << 1;
  const float* arow = x + (size_t)(row0 + n) * 64;
  v8f c0{}, c1{}, c2{}, c3{};
  gemm_strip64(arow, B, H_, 64, koff, n, c0, c1, c2, c3);
  int half = lane >> 4;
  float* crow = P + (size_t)row0 * 256 + g * 64 + n;
#pragma unroll
  for (int v = 0; v < 8; ++v) {
    float* d = crow + (size_t)(v + 8 * half) * 256;
    d[0]  = c0[v];
    d[16] = c1[v];
    d[32] = c2[v];
    d[48] = c3[v];
  }
}

// tiny per-layer tables:
//   G2[g,j] = gf0[g]@psi_w[129:193] + psi_b   (50x128)
//   G3[g,j] = gf0[g]@phi_w[192:256] + phi_b   (50x64)
__global__ void k_G(const float* __restrict__ gf0,
                    const float* __restrict__ psiWl, const float* __restrict__ psiBl,
                    const float* __restrict__ phiWl, const float* __restrict__ phiBl,
                    float* __restrict__ G2, float* __restrict__ G3) {
  int t = blockIdx.x * blockDim.x + threadIdx.x;
  if (t < NG_ * H_) {
    int g = t >> 7, j = t & 127;
    const float* gr = gf0 + g * 64;
    float s = psiBl[j];
    for (int k = 0; k < 64; ++k) s += gr[k] * psiWl[(size_t)(129 + k) * H_ + j];
    G2[t] = s;
  } else if (t < NG_ * H_ + NG_ * 64) {
    int t2 = t - NG_ * H_;
    int g = t2 >> 6, j = t2 & 63;
    const float* gr = gf0 + g * 64;
    float s = phiBl[j];
    for (int k = 0; k < 64; ++k) s += gr[k] * phiWl[(size_t)(192 + k) * 64 + j];
    G3[t2] = s;
  }
}

// ---------------------------------------------------------------------------
// CSR gather (per layer): one wave per node, lane owns a float4 column slice.
// agg[v] = deg>0 ? P1[v] + G2[b(v)] + (sum P2[src] + (sum ea)*w3)/deg : 0
// Replaces scatter-atomics + memset + separate agg pass; zero float atomics.
// ---------------------------------------------------------------------------
__global__ void k_gather(const float* __restrict__ P, const int* __restrict__ off,
                         const int* __restrict__ deg_i, const int* __restrict__ elist,
                         const float* __restrict__ eaval, const float* __restrict__ G2,
                         const int* __restrict__ batch, const float* __restrict__ psiWl,
                         float* __restrict__ agg) {
  int wid  = (blockIdx.x * blockDim.x + threadIdx.x) >> 5;
  int lane = threadIdx.x & 31;
  if (wid >= N_NODES) return;
  int v = wid;
  int o = off[v];
  int d = deg_i[v];
  float4 acc = make_float4(0.f, 0.f, 0.f, 0.f);
  float  es  = 0.f;
  for (int i = 0; i < d; ++i) {
    int s = elist[o + i];
    const float4 p2 = *(const float4*)(P + (size_t)s * 256 + 128 + lane * 4);
    acc.x += p2.x; acc.y += p2.y; acc.z += p2.z; acc.w += p2.w;
    es += eaval[o + i];
  }
  float4 outv = make_float4(0.f, 0.f, 0.f, 0.f);
  if (d > 0) {
    float inv = 1.0f / (float)d;
    const float4 p1 = *(const float4*)(P + (size_t)v * 256 + lane * 4);
    const float4 g2 = *(const float4*)(G2 + (size_t)batch[v] * H_ + lane * 4);
    const float4 w3 = *(const float4*)(psiWl + (size_t)128 * H_ + lane * 4);
    outv.x = p1.x + g2.x + (acc.x + es * w3.x) * inv;
    outv.y = p1.y + g2.y + (acc.y + es * w3.y) * inv;
    outv.z = p1.z + g2.z + (acc.z + es * w3.z) * inv;
    outv.w = p1.w + g2.w + (acc.w + es * w3.w) * inv;
  }
  *(float4*)(agg + (size_t)v * H_ + lane * 4) = outv;
}

// xnew = agg@F1 + x@F2 + G3[b(v)];  x = (l==0) ? relu(xnew) : x + relu(xnew)  [WMMA]
__global__ void k_phi(const float* __restrict__ agg, float* xbuf,
                      const float* __restrict__ phiWl, const float* __restrict__ G3,
                      const int* __restrict__ batch, int layer) {
  int wid  = (blockIdx.x * blockDim.x + threadIdx.x) >> 5;
  int lane = threadIdx.x & 31;
  if (wid >= NTILES) return;
  int row0 = wid * 16;
  int n    = lane & 15;
  int koff = (lane >> 4) << 1;
  v8f c0{}, c1{}, c2{}, c3{};
  gemm_strip64(agg + (size_t)(row0 + n) * H_, phiWl, 64, H_, koff, n, c0, c1, c2, c3);
  gemm_strip64(xbuf + (size_t)(row0 + n) * 64, phiWl + (size_t)H_ * 64, 64, 64,
               koff, n, c0, c1, c2, c3);
  int half = lane >> 4;
#pragma unroll
  for (int v = 0; v < 8; ++v) {
    int row = row0 + v + 8 * half;
    const float* g3 = G3 + (size_t)batch[row] * 64;
    float* xr = xbuf + (size_t)row * 64;
    float r0 = fmaxf(c0[v] + g3[n],      0.f);
    float r1 = fmaxf(c1[v] + g3[16 + n], 0.f);
    float r2 = fmaxf(c2[v] + g3[32 + n], 0.f);
    float r3 = fmaxf(c3[v] + g3[48 + n], 0.f);
    if (layer == 0) { xr[n] = r0; xr[16+n] = r1; xr[32+n] = r2; xr[48+n] = r3; }
    else           { xr[n] += r0; xr[16+n] += r1; xr[32+n] += r2; xr[48+n] += r3; }
  }
}

// head: out = x@lin_w + lin_b ; mu = out0 ; sigma = softplus(out1)
__global__ void k_out(const float* __restrict__ x, const float* __restrict__ lw,
                      const float* __restrict__ lb, float* __restrict__ out) {
  int nidx = blockIdx.x * blockDim.x + threadIdx.x;
  if (nidx >= N_NODES) return;
  const float* xr = x + (size_t)nidx * 64;
  float a = 0.f, b = 0.f;
  for (int k = 0; k < 64; ++k) {
    float xv = xr[k];
    a += xv * lw[2 * k];
    b += xv * lw[2 * k + 1];
  }
  a += lb[0]; b += lb[1];
  out[nidx]           = a;
  out[N_NODES + nidx] = (b > 20.f) ? b : log1pf(expf(b));
}

// ---------------------------------------------------------------------------
extern "C" void kernel_launch(void* const* d_in, const int* in_sizes, int n_in,
                              void* d_out, int out_size, void* d_ws, size_t ws_size,
                              hipStream_t stream) {
  (void)in_sizes; (void)n_in; (void)out_size; (void)ws_size;
  const float* x_feats = (const float*)d_in[0];
  const int*   station = (const int*)d_in[1];
  const int*   eidx    = (const int*)d_in[2];
  const int*   src     = eidx;
  const int*   dst     = eidx + E_EDGES;
  const float* eattr   = (const float*)d_in[3];
  const int*   batch   = (const int*)d_in[4];
  const float* emb     = (const float*)d_in[5];
  const float* egamma  = (const float*)d_in[6];
  const float* ebeta   = (const float*)d_in[7];
  const float* psi_w   = (const float*)d_in[8];
  const float* psi_b   = (const float*)d_in[9];
  const float* phi_w   = (const float*)d_in[10];
  const float* phi_b   = (const float*)d_in[11];
  // d_in[12], d_in[13] (rho_w, rho_b): dead code — output depends only on x
  const float* lin_w   = (const float*)d_in[14];
  const float* lin_b   = (const float*)d_in[15];
  float* out = (float*)d_out;

  float* ws    = (float*)d_ws;
  float* x     = ws;                                 // N*64
  float* P     = x     + (size_t)N_NODES * 64;       // N*256  (P1 | P2)
  float* agg   = P     + (size_t)N_NODES * 256;      // N*128
  float* eaval = agg   + (size_t)N_NODES * 128;      // E  (bucketed norm attr)
  float* gf0   = eaval + E_EDGES;                    // NG*64
  float* G2    = gf0   + NG_ * 64;                   // NG*128
  float* G3    = G2    + NG_ * H_;                   // NG*64
  float* stats = G3    + NG_ * 64;                   // 8
  float* part  = stats + 8;                          // 512
  int*   deg_i = (int*)(part + 512);                 // N
  int*   off   = deg_i + N_NODES;                    // N
  int*   cursor= off   + N_NODES;                    // N
  int*   elist = cursor+ N_NODES;                    // E  (bucketed src nodes)
  int*   bsum  = elist + E_EDGES;                    // 256

  const int EB = (E_EDGES + 255) / 256;

  // edge-attr layernorm stats
  k_stat1<<<256, 256, 0, stream>>>(eattr, part);
  k_stat2<<<1, 256, 0, stream>>>(part, stats);

  // CSR build (layer-invariant)
  hipMemsetAsync(deg_i,  0, N_NODES * sizeof(int), stream);
  hipMemsetAsync(cursor, 0, N_NODES * sizeof(int), stream);
  k_count<<<EB, 256, 0, stream>>>(dst, deg_i);
  k_scan1<<<NSCAN, 256, 0, stream>>>(deg_i, off, bsum);
  k_scan2<<<1, 256, 0, stream>>>(bsum);
  k_scan3<<<NSCAN, 256, 0, stream>>>(off, bsum);
  k_fill<<<EB, 256, 0, stream>>>(src, dst, eattr, stats, egamma, ebeta,
                                 off, cursor, elist, eaval);

  // node features + graph pooling
  k_buildx<<<(N_NODES * 64 + 255) / 256, 256, 0, stream>>>(x_feats, station, emb, x);
  k_gf0<<<NG_, 256, 0, stream>>>(x, gf0);

  for (int l = 0; l < L_; ++l) {
    const float* psiWl = psi_w + (size_t)l * PSI_IN * H_;
    const float* psiBl = psi_b + (size_t)l * H_;
    const float* phiWl = phi_w + (size_t)l * PHI_IN * 64;
    const float* phiBl = phi_b + (size_t)l * 64;

    k_p12<<<(NTILES * 4 * 32 + 255) / 256, 256, 0, stream>>>(x, psiWl, P);
    k_G<<<(NG_ * 192 + 255) / 256, 256, 0, stream>>>(gf0, psiWl, psiBl,
                                                     phiWl, phiBl, G2, G3);
    k_gather<<<(N_NODES * 32 + 255) / 256, 256, 0, stream>>>(P, off, deg_i, elist,
                                                             eaval, G2, batch,
                                                             psiWl, agg);
    k_phi<<<(NTILES * 32 + 255) / 256, 256, 0, stream>>>(agg, x, phiWl, G3,
                                                         batch, l);
  }
  k_out<<<(N_NODES + 255) / 256, 256, 0, stream>>>(x, lin_w, lin_b, out);
}